// HDLoss_47648367181933
// MI455X (gfx1250) — compile-verified
//
#include <hip/hip_runtime.h>
#include <hip/hip_bf16.h>

typedef __attribute__((ext_vector_type(2))) float v2f;
typedef __attribute__((ext_vector_type(8))) float v8f;

// ---------------------------------------------------------------------------
// Pass 1: per-64x64-block sums of |n-a| (-> blk) and |a-p| (-> sap).
// Memory-bound: 151 MB of reads; float4 (b128) coalesced loads, wave32
// shfl reductions. One 256-thread workgroup per block; 3072 workgroups.
// ---------------------------------------------------------------------------
__global__ __launch_bounds__(256) void hd_block_sums(
    const float* __restrict__ a, const float* __restrict__ p,
    const float* __restrict__ n, float* __restrict__ blk,
    float* __restrict__ sap)
{
    const int t   = blockIdx.x;          // 0..3071 = img*64 + bi*8 + bj
    const int img = t >> 6;
    const int tij = t & 63;
    const int bi  = tij >> 3, bj = tij & 7;
    const size_t base = (size_t)img * (512u * 512u)
                      + (size_t)bi * (64u * 512u) + (size_t)bj * 64u;
    const int tid = threadIdx.x;

    float s1 = 0.f, s2 = 0.f;
#pragma unroll
    for (int k = 0; k < 4; ++k) {
        const int q    = tid + k * 256;      // float4 index inside tile
        const int row  = q >> 4;             // 16 float4 per 64-wide row
        const int col4 = q & 15;
        const size_t off = base + (size_t)row * 512u + (size_t)col4 * 4u;
        const float4 va = *(const float4*)(a + off);
        const float4 vp = *(const float4*)(p + off);
        const float4 vn = *(const float4*)(n + off);
        s1 += fabsf(vn.x - va.x) + fabsf(vn.y - va.y)
            + fabsf(vn.z - va.z) + fabsf(vn.w - va.w);
        s2 += fabsf(va.x - vp.x) + fabsf(va.y - vp.y)
            + fabsf(va.z - vp.z) + fabsf(va.w - vp.w);
    }
    // wave32 reduction
#pragma unroll
    for (int m = 16; m > 0; m >>= 1) {
        s1 += __shfl_xor(s1, m, 32);
        s2 += __shfl_xor(s2, m, 32);
    }
    __shared__ float r1[8], r2[8];
    const int wave = tid >> 5, lane = tid & 31;
    if (lane == 0) { r1[wave] = s1; r2[wave] = s2; }
    __syncthreads();
    if (tid == 0) {
        float t1 = 0.f, t2 = 0.f;
#pragma unroll
        for (int w = 0; w < 8; ++w) { t1 += r1[w]; t2 += r2[w]; }
        blk[t] = t1;
        sap[t] = t2;
    }
}

// ---------------------------------------------------------------------------
// R matrix (symmetric, tridiagonal: diag [1,2,2,2,2,2,2,1], off-diag 1),
// block-diagonal doubled to 16x16 (two images per wave).
// ---------------------------------------------------------------------------
__device__ __forceinline__ float R2v(int k, int nn) {
    const int ki = k & 7, ni = nn & 7;
    const bool same = ((k >> 3) == (nn >> 3));
    int d = ki - ni; d = d < 0 ? -d : d;
    const float v = (d == 0) ? (((ki == 0) || (ki == 7)) ? 1.f : 2.f)
                             : ((d == 1) ? 1.f : 0.f);
    return same ? v : 0.f;
}

__device__ __forceinline__ float blk2v(const float* B0, const float* B1,
                                       int m, int k) {
    const bool lo = (m < 8) && (k < 8);
    const bool hi = (m >= 8) && (k >= 8);
    const float v0 = B0[(m & 7) * 8 + (k & 7)];
    const float v1 = B1[(m & 7) * 8 + (k & 7)];
    return lo ? v0 : (hi ? v1 : 0.f);
}

__device__ __forceinline__ float cw(int i) {
    return ((i == 0) || (i == 7)) ? 1.f : 2.f;
}

// ---------------------------------------------------------------------------
// Pass 2: mb = (R * blk * R) / (c_i*c_j*diff); partial loss = sum mb*sap.
// Two images per wave as a block-diagonal 16x16 f32 matmul, done with
// chains of v_wmma_f32_16x16x4_f32 (K=4 slices, exact f32).
//
// WMMA lane layouts used (wave32):
//   A 16x4 f32 : lane L holds M=L%16; v0 = K-slice +0 (L<16) / +2 (L>=16);
//                v1 = +1 / +3.                          [ISA 7.12.2, documented]
//   C/D 16x16  : VGPR r holds M=r (lanes 0-15) and M=r+8 (lanes 16-31),
//                N = lane%16.                           [ISA 7.12.2, documented]
//   B 4x16     : assumed mirror of A/C convention: v0 = row K+0 (lanes 0-15)
//                / K+2 (lanes 16-31); v1 = K+1 / K+3.
// ---------------------------------------------------------------------------
__global__ __launch_bounds__(32) void hd_mask_loss(
    const float* __restrict__ blk, const float* __restrict__ sap,
    float* __restrict__ part)
{
    const int g    = blockIdx.x;     // 0..23
    const int lane = threadIdx.x;    // 0..31
    const int img0 = 2 * g, img1 = 2 * g + 1;

    __shared__ float B0[64], B1[64], Tl[256];
    B0[lane]      = blk[img0 * 64 + lane];
    B0[lane + 32] = blk[img0 * 64 + lane + 32];
    B1[lane]      = blk[img1 * 64 + lane];
    B1[lane + 32] = blk[img1 * 64 + lane + 32];
    __syncthreads();

    float diff0 = 0.f, diff1 = 0.f;
#pragma unroll
    for (int i = 0; i < 64; ++i) { diff0 += B0[i]; diff1 += B1[i]; }

    const int M    = lane & 15;
    const int hi2  = (lane >= 16) ? 2 : 0;
    const int mof  = (lane >= 16) ? 8 : 0;

    // ---- T = blk2 * R2 ----
    v2f Aa[4], Bb[4];
#pragma unroll
    for (int kk = 0; kk < 4; ++kk) {
        const int K0 = 4 * kk + hi2;
        Aa[kk].x = blk2v(B0, B1, M, K0);
        Aa[kk].y = blk2v(B0, B1, M, K0 + 1);
        Bb[kk].x = R2v(K0, M);
        Bb[kk].y = R2v(K0 + 1, M);
    }
    v8f acc = {};
#pragma unroll
    for (int kk = 0; kk < 4; ++kk) {
        acc = __builtin_amdgcn_wmma_f32_16x16x4_f32(
            false, Aa[kk], false, Bb[kk], (short)0, acc, false, false);
    }

    // spill T (D layout) to LDS as row-major T[m][n]
#pragma unroll
    for (int r = 0; r < 8; ++r) Tl[(r + mof) * 16 + M] = acc[r];
    __syncthreads();

    // ---- mb2 = R2 * T ----
    v2f Ac[4], Bc[4];
#pragma unroll
    for (int kk = 0; kk < 4; ++kk) {
        const int K0 = 4 * kk + hi2;
        Ac[kk].x = R2v(M, K0);
        Ac[kk].y = R2v(M, K0 + 1);
        Bc[kk].x = Tl[K0 * 16 + M];
        Bc[kk].y = Tl[(K0 + 1) * 16 + M];
    }
    v8f acc2 = {};
#pragma unroll
    for (int kk = 0; kk < 4; ++kk) {
        acc2 = __builtin_amdgcn_wmma_f32_16x16x4_f32(
            false, Ac[kk], false, Bc[kk], (short)0, acc2, false, false);
    }

    // ---- loss contribution: sum over valid (m,n) of mb * sap ----
    float contrib = 0.f;
    const bool hi = (lane >= 16);
#pragma unroll
    for (int r = 0; r < 8; ++r) {
        const int m = r + mof;           // row in 16x16
        const float v = acc2[r];         // mb2[m][M]
        if (!hi) {
            if (M < 8)
                contrib += v * sap[img0 * 64 + m * 8 + M]
                         / (cw(m) * cw(M) * diff0);
        } else {
            if (M >= 8) {
                const int i = m - 8, j = M - 8;
                contrib += v * sap[img1 * 64 + i * 8 + j]
                         / (cw(i) * cw(j) * diff1);
            }
        }
    }
#pragma unroll
    for (int s = 16; s > 0; s >>= 1) contrib += __shfl_xor(contrib, s, 32);
    if (lane == 0) part[g] = contrib;
}

// ---------------------------------------------------------------------------
// Pass 3: reduce 24 partials -> scalar mean.
// ---------------------------------------------------------------------------
__global__ __launch_bounds__(32) void hd_finalize(
    const float* __restrict__ part, float* __restrict__ out)
{
    const int lane = threadIdx.x;
    float v = (lane < 24) ? part[lane] : 0.f;
#pragma unroll
    for (int s = 16; s > 0; s >>= 1) v += __shfl_xor(v, s, 32);
    if (lane == 0) out[0] = v / 12582912.0f;   // 16*3*512*512
}

extern "C" void kernel_launch(void* const* d_in, const int* in_sizes, int n_in,
                              void* d_out, int out_size, void* d_ws, size_t ws_size,
                              hipStream_t stream) {
    const float* a = (const float*)d_in[0];
    const float* p = (const float*)d_in[1];
    const float* n = (const float*)d_in[2];
    float* ws   = (float*)d_ws;
    float* blk  = ws;            // 3072 floats
    float* sap  = ws + 3072;     // 3072 floats
    float* part = ws + 6144;     // 24 floats

    hd_block_sums<<<3072, 256, 0, stream>>>(a, p, n, blk, sap);
    hd_mask_loss<<<24, 32, 0, stream>>>(blk, sap, part);
    hd_finalize<<<1, 32, 0, stream>>>(part, (float*)d_out);
}